// QuadricGrid_74139725464054
// MI455X (gfx1250) — compile-verified
//
#include <hip/hip_runtime.h>
#include <hip/hip_bf16.h>

#define RESO 128
#define QEPS 1e-8f

// Fused quadric SDF + normal kernel.
//  - Stages xLayer/yLayer/zLayer (3*128 floats) into LDS via CDNA5 async
//    global->LDS b128 copies (one wave-wide copy per layer), synced with
//    s_wait_asynccnt.
//  - Grid-stride loop; each iteration handles one sdf element and one
//    render element (both lists are the same length in practice).
__global__ __launch_bounds__(256) void quadric_grid_kernel(
    const float* __restrict__ renderPts,   // [Nr,3]
    const int*   __restrict__ renderIdx,   // [Nr]
    const float* __restrict__ sdfPts,      // [Ns,3]
    const int*   __restrict__ sdfIdx,      // [Ns]
    const float* __restrict__ xL,          // [128]
    const float* __restrict__ yL,          // [128]
    const float* __restrict__ zL,          // [128]
    const float* __restrict__ offs,        // [4]
    float* __restrict__ outSdf,            // [Ns]
    float* __restrict__ outNrm,            // [Nr,3]
    int Ns, int Nr)
{
    __shared__ float lut[3 * RESO];  // [0:128)=x, [128:256)=y, [256:384)=z

    const int t = threadIdx.x;
    const int w = t >> 5;      // wave id within block (wave32)
    const int l = t & 31;      // lane id

    // --- Async fill of the LDS lookup tables -------------------------------
    // Waves 0,1,2 each copy one 512-byte layer: lane l moves 16 bytes
    // (floats 4l..4l+3). EXEC is all-ones within each issuing wave as
    // required. hipMalloc allocations are >=256B aligned, so b128 is legal.
    if (w < 3) {
        const float* src = (w == 0) ? xL : (w == 1) ? yL : zL;
        unsigned lds_addr =
            (unsigned)(uintptr_t)(&lut[w * RESO + 4 * l]);
        unsigned long long gaddr =
            (unsigned long long)(uintptr_t)(src + 4 * l);
        asm volatile("global_load_async_to_lds_b128 %0, %1, off"
                     :
                     : "v"(lds_addr), "v"(gaddr)
                     : "memory");
    }
    // Wait for this wave's async copies (no-op for waves 3..7), then make
    // all waves' LDS writes visible block-wide.
    asm volatile("s_wait_asynccnt 0" ::: "memory");
    __syncthreads();

    // Uniform linear/constant coefficients (scalar loads).
    const float b0 = offs[0];
    const float b1 = offs[1];
    const float b2 = offs[2];
    const float cc = offs[3];

    const int stride = (int)(gridDim.x * blockDim.x);
    const int nmax   = (Ns > Nr) ? Ns : Nr;

    for (int i = (int)(blockIdx.x * blockDim.x) + t; i < nmax; i += stride) {
        // ---------------- SDF list ----------------
        if (i < Ns) {
            const int idx = sdfIdx[i];
            const float ax = lut[(idx >> 14) & (RESO - 1)];
            const float ay = lut[RESO + ((idx >> 7) & (RESO - 1))];
            const float az = lut[2 * RESO + (idx & (RESO - 1))];
            const float x = sdfPts[3 * i + 0];
            const float y = sdfPts[3 * i + 1];
            const float z = sdfPts[3 * i + 2];
            const float s = ax * x * x + ay * y * y + az * z * z
                          + b0 * x + b1 * y + b2 * z + cc;
            outSdf[i] = s;
        }
        // ---------------- Normal list ----------------
        if (i < Nr) {
            const int idx = renderIdx[i];
            const float ax = lut[(idx >> 14) & (RESO - 1)];
            const float ay = lut[RESO + ((idx >> 7) & (RESO - 1))];
            const float az = lut[2 * RESO + (idx & (RESO - 1))];
            const float x = renderPts[3 * i + 0];
            const float y = renderPts[3 * i + 1];
            const float z = renderPts[3 * i + 2];
            const float gx = 2.0f * ax * x + b0;
            const float gy = 2.0f * ay * y + b1;
            const float gz = 2.0f * az * z + b2;
            const float d  = gx * gx + gy * gy + gz * gz;
            const float inv = 1.0f / (sqrtf(d) + QEPS);
            outNrm[3 * i + 0] = gx * inv;
            outNrm[3 * i + 1] = gy * inv;
            outNrm[3 * i + 2] = gz * inv;
        }
    }
}

extern "C" void kernel_launch(void* const* d_in, const int* in_sizes, int n_in,
                              void* d_out, int out_size, void* d_ws, size_t ws_size,
                              hipStream_t stream) {
    // Input order (setup_inputs): renderPointList, renderIndexList,
    // sdfPointList, sdfIndexList, xLayer, yLayer, zLayer, offset.
    const float* renderPts = (const float*)d_in[0];
    const int*   renderIdx = (const int*)  d_in[1];
    const float* sdfPts    = (const float*)d_in[2];
    const int*   sdfIdx    = (const int*)  d_in[3];
    const float* xL        = (const float*)d_in[4];
    const float* yL        = (const float*)d_in[5];
    const float* zL        = (const float*)d_in[6];
    const float* offs      = (const float*)d_in[7];

    const int Nr = in_sizes[1];
    const int Ns = in_sizes[3];

    float* outSdf = (float*)d_out;       // [Ns]
    float* outNrm = outSdf + Ns;         // [Nr,3] flat

    const int nmax    = (Ns > Nr) ? Ns : Nr;
    const int block   = 256;             // 8 wave32s
    int blocks        = (nmax + block - 1) / block;
    if (blocks > 4096) blocks = 4096;    // persistent-ish grid, amortize LDS fill
    if (blocks < 1) blocks = 1;

    quadric_grid_kernel<<<blocks, block, 0, stream>>>(
        renderPts, renderIdx, sdfPts, sdfIdx, xL, yL, zL, offs,
        outSdf, outNrm, Ns, Nr);
}